// ReversePolicy_82454782149150
// MI455X (gfx1250) — compile-verified
//
#include <hip/hip_runtime.h>
#include <hip/hip_bf16.h>
#include <cstddef>
#include <cstdint>

typedef float v2f __attribute__((ext_vector_type(2)));
typedef float v8f __attribute__((ext_vector_type(8)));

#define NEGINF (-__builtin_inff())

// Model dims fixed by the reference
#define NN 256   // nodes
#define DD 128   // node dim
#define EE 64    // edge dim
#define FF 320   // 2*DD + EE
#define FSTRIDE 324  // padded LDS row stride (floats): 324%64==4 -> conflict-free rows

#ifndef __has_builtin
#define __has_builtin(x) 0
#endif

#if __has_builtin(__builtin_amdgcn_global_load_async_to_lds_b128)
#define USE_ASYNC_LDS 1
// The builtin's params are pointers to 16-byte int vectors:
//   param0: global (as1) pointer, param1: LDS (as3) pointer
typedef int v4i_vs __attribute__((vector_size(16)));
typedef __attribute__((address_space(1))) v4i_vs as1_v4i;
typedef __attribute__((address_space(3))) v4i_vs as3_v4i;
#else
#define USE_ASYNC_LDS 0
#endif

// ---------------------------------------------------------------------------
// Edit logits: one wave (32 threads) per (b, k-tile of 16).
// Stage 16 feature rows (320 f32) into LDS (async global->LDS B128 when the
// toolchain exposes it), then 80 chained V_WMMA_F32_16X16X4_F32 computing
// G[k, c] for all 4 weight columns, then select column b_idx[k].
// ---------------------------------------------------------------------------
__global__ void edit_logits_kernel(const float* __restrict__ hN,
                                   const float* __restrict__ hE,
                                   const float* __restrict__ editW,   // [320,4] row-major
                                   const float* __restrict__ editB,   // [4]
                                   const int* __restrict__ iIdx,
                                   const int* __restrict__ jIdx,
                                   const int* __restrict__ bIdx,
                                   const unsigned char* __restrict__ editFeas, // [B,K]
                                   float* __restrict__ out,
                                   int K, int outStride) {
    const int lane = threadIdx.x;         // 0..31, single wave per block
    const int b    = blockIdx.y;
    const int k0   = blockIdx.x * 16;

    __shared__ float feat[16 * FSTRIDE];  // 16 feature rows, padded
    __shared__ float wlds[FF * 4];        // edit_W staged
    __shared__ float glds[16 * 16];       // D-matrix spill for column select

    // Stage edit_W (1280 floats)
    for (int t = lane; t < FF * 4; t += 32) wlds[t] = editW[t];

    // Stage 16 rows x 80 float4-chunks, fully coalesced per-row segments
    for (int t = lane; t < 16 * 80; t += 32) {
        const int r = t / 80;
        const int c = t % 80;             // chunk of 4 floats
        const int k = k0 + r;
        float* dst = &feat[r * FSTRIDE + c * 4];
        if (k < K) {
            const int i = iIdx[k];
            const int j = jIdx[k];
            const int f = c * 4;
            const float* src;
            if (f < DD)            src = hN + ((size_t)(b * NN + i)) * DD + f;
            else if (f < 2 * DD)   src = hN + ((size_t)(b * NN + j)) * DD + (f - DD);
            else                   src = hE + (((size_t)(b * NN + i)) * NN + j) * EE + (f - 2 * DD);
#if USE_ASYNC_LDS
            // CDNA5 direct global->LDS async copy (ASYNCcnt-tracked, no VGPR round trip)
            __builtin_amdgcn_global_load_async_to_lds_b128(
                (as1_v4i*)(uintptr_t)src, (as3_v4i*)dst, 0, 0);
#else
            *(float4*)dst = *(const float4*)src;
#endif
        } else {
#if !USE_ASYNC_LDS
            *(float4*)dst = make_float4(0.f, 0.f, 0.f, 0.f);
#endif
            // async path: tail rows stay uninitialized; WMMA output rows are
            // independent and tail rows are never written out (K%16==0 anyway)
        }
    }
#if USE_ASYNC_LDS
    asm volatile("s_wait_asynccnt 0" ::: "memory");
#endif
    __syncthreads();

    // A-fragment per ISA: lanes 0-15 hold M=0..15 with K={0,1}; lanes 16-31 K={2,3}
    const int row  = lane & 15;
    const int hi   = lane >> 4;           // 0 or 1
    const int base = hi * 2;              // K offset within group of 4
    const int col  = lane & 15;           // B/D column
    const int colc = col & 3;             // clamped weight column (broadcast read)
    const float msk = (col < 4) ? 1.0f : 0.0f;  // zero the 12 unused N columns

    const float* fr = &feat[row * FSTRIDE + base];
    const float* wr = &wlds[base * 4 + colc];

    v8f acc = {};
#pragma unroll 4
    for (int s = 0; s < 80; ++s) {
        v2f a;
        a.x = fr[4 * s];
        a.y = fr[4 * s + 1];
        v2f bf;
        bf.x = wr[16 * s]     * msk;      // W[4s+base  , colc]
        bf.y = wr[16 * s + 4] * msk;      // W[4s+base+1, colc]
        acc = __builtin_amdgcn_wmma_f32_16x16x4_f32(
            /*neg_a=*/false, a, /*neg_b=*/false, bf,
            /*c_mod=*/(short)0, acc, /*reuse_a=*/false, /*reuse_b=*/false);
    }

    // D layout: lane L, VGPR v -> element (M = v + 8*(L>=16), N = L%16)
#pragma unroll
    for (int v = 0; v < 8; ++v) {
        glds[(v + 8 * hi) * 16 + col] = acc[v];
    }
    __syncthreads();

    if (lane < 16) {
        const int k = k0 + lane;
        if (k < K) {
            const int c = bIdx[k];
            const float val = glds[lane * 16 + c] + editB[c];
            const unsigned char f = editFeas[(size_t)b * K + k];
            out[(size_t)b * outStride + k] = f ? val : NEGINF;
        }
    }
}

// ---------------------------------------------------------------------------
// Precompute aterm[a] = dot(atom_embed[a], insert_W[128:256]) for 16 atoms.
// One wave per atom.
// ---------------------------------------------------------------------------
__global__ void aterm_kernel(const float* __restrict__ atomEmbed,
                             const float* __restrict__ insW,
                             float* __restrict__ aterm) {
    const int lane = threadIdx.x & 31;
    const int a    = threadIdx.x >> 5;
    float p = 0.f;
#pragma unroll
    for (int d = lane; d < DD; d += 32)
        p += atomEmbed[a * DD + d] * insW[DD + d];
    p += __shfl_xor(p, 16, 32);
    p += __shfl_xor(p, 8, 32);
    p += __shfl_xor(p, 4, 32);
    p += __shfl_xor(p, 2, 32);
    p += __shfl_xor(p, 1, 32);
    if (lane == 0) aterm[a] = p;
}

// ---------------------------------------------------------------------------
// Insert logits: one wave per (b,k). Wave cooperatively loads the 512B node
// row (float4 per lane), dots with insert_W[:128], wave-reduces via shuffles.
// ---------------------------------------------------------------------------
__global__ void insert_logits_kernel(const float* __restrict__ hN,
                                     const float* __restrict__ insW,
                                     const float* __restrict__ insB,
                                     const int* __restrict__ siteIdx,
                                     const int* __restrict__ atomIdx,
                                     const unsigned char* __restrict__ feas, // [B,K]
                                     const float* __restrict__ aterm,
                                     float* __restrict__ out,
                                     int K, int editOff, int outStride, int total) {
    const int lane = threadIdx.x & 31;
    const int w    = threadIdx.x >> 5;
    const int task = blockIdx.x * (blockDim.x >> 5) + w;
    if (task >= total) return;                   // whole wave exits together
    const int b = task / K;
    const int k = task - b * K;

    const int site = siteIdx[k];
    const float* rowp = hN + ((size_t)b * NN + site) * DD;
    const float4 hv = *(const float4*)(rowp + lane * 4);
    const float4 wv = *(const float4*)(insW + lane * 4);
    float p = hv.x * wv.x + hv.y * wv.y + hv.z * wv.z + hv.w * wv.w;
    p += __shfl_xor(p, 16, 32);
    p += __shfl_xor(p, 8, 32);
    p += __shfl_xor(p, 4, 32);
    p += __shfl_xor(p, 2, 32);
    p += __shfl_xor(p, 1, 32);
    if (lane == 0) {
        const float val = p + aterm[atomIdx[k]] + insB[0];
        const unsigned char f = feas[(size_t)b * K + k];
        out[(size_t)b * outStride + editOff + k] = f ? val : NEGINF;
    }
}

// ---------------------------------------------------------------------------
// Stop logit: one block per batch; thread d sums over the node axis
// (coalesced in d), multiplies by stop_W[d], block-reduces.
// ---------------------------------------------------------------------------
__global__ void stop_kernel(const float* __restrict__ hN,
                            const float* __restrict__ stopW,
                            const float* __restrict__ stopB,
                            const unsigned char* __restrict__ feas,
                            float* __restrict__ out,
                            int outStride) {
    const int b = blockIdx.x;
    const int d = threadIdx.x;   // 0..127
    const float* base = hN + (size_t)b * NN * DD;
    float s = 0.f;
    for (int n = 0; n < NN; ++n) s += base[n * DD + d];
    __shared__ float red[DD];
    red[d] = s * stopW[d];
    __syncthreads();
    for (int off = DD / 2; off >= 1; off >>= 1) {
        if (d < off) red[d] += red[d + off];
        __syncthreads();
    }
    if (d == 0) {
        const float val = red[0] * (1.0f / NN) + stopB[0];
        out[(size_t)b * outStride + (outStride - 1)] = feas[b] ? val : NEGINF;
    }
}

extern "C" void kernel_launch(void* const* d_in, const int* in_sizes, int n_in,
                              void* d_out, int out_size, void* d_ws, size_t ws_size,
                              hipStream_t stream) {
    const float* hN        = (const float*)d_in[0];
    const float* hE        = (const float*)d_in[1];
    const float* editW     = (const float*)d_in[2];
    const float* editB     = (const float*)d_in[3];
    const float* atomEmbed = (const float*)d_in[4];
    const float* insW      = (const float*)d_in[5];
    const float* insB      = (const float*)d_in[6];
    const float* stopW     = (const float*)d_in[7];
    const float* stopB     = (const float*)d_in[8];
    const int* iIdx        = (const int*)d_in[9];
    const int* jIdx        = (const int*)d_in[10];
    const int* bIdx        = (const int*)d_in[11];
    const int* siteIdx     = (const int*)d_in[12];
    const int* atomIdx     = (const int*)d_in[13];
    const unsigned char* editFeas = (const unsigned char*)d_in[14];
    const unsigned char* insFeas  = (const unsigned char*)d_in[15];
    const unsigned char* stopFeas = (const unsigned char*)d_in[16];
    float* out = (float*)d_out;

    const int K_EDIT = in_sizes[9];
    const int K_INS  = in_sizes[12];
    const int B      = in_sizes[16];
    const int outStride = K_EDIT + K_INS + 1;
    const int nAtoms = in_sizes[4] / DD;

    float* aterm = (float*)d_ws;

    // 1) atom-term table (16 dot products)
    aterm_kernel<<<1, nAtoms * 32, 0, stream>>>(atomEmbed, insW, aterm);

    // 2) edit logits via f32 WMMA
    dim3 egrid((K_EDIT + 15) / 16, B);
    edit_logits_kernel<<<egrid, 32, 0, stream>>>(hN, hE, editW, editB, iIdx, jIdx,
                                                 bIdx, editFeas, out, K_EDIT, outStride);

    // 3) insert logits: one wave per (b,k)
    const int total = B * K_INS;
    const int wavesPerBlock = 8;  // 256 threads
    const int iblocks = (total + wavesPerBlock - 1) / wavesPerBlock;
    insert_logits_kernel<<<iblocks, wavesPerBlock * 32, 0, stream>>>(
        hN, insW, insB, siteIdx, atomIdx, insFeas, aterm, out,
        K_INS, K_EDIT, outStride, total);

    // 4) stop logit
    stop_kernel<<<B, DD, 0, stream>>>(hN, stopW, stopB, stopFeas, out, outStride);
}